// FPSsubsample_18004502904910
// MI455X (gfx1250) — compile-verified
//
#include <hip/hip_runtime.h>
#include <hip/hip_bf16.h>
#include <stdint.h>

#define FPS_BLOCK 1024
#define MAX_N 2048

// ---- CDNA5 async global->LDS copy (tracked by ASYNCcnt) ----
__device__ __forceinline__ void async_copy_b128(uint32_t lds_addr, uint32_t voff,
                                                const void* saddr_base) {
  // GVS mode: mem = SADDR(64) + VADDR(32); VDST = LDS byte address
  asm volatile("global_load_async_to_lds_b128 %0, %1, %2"
               :: "v"(lds_addr), "v"(voff), "s"(saddr_base)
               : "memory");
}
__device__ __forceinline__ void wait_async0() {
  asm volatile("s_wait_asynccnt 0" ::: "memory");
}

__device__ __forceinline__ void combine_max_first(float& v, int& j, float ov, int oj) {
  if (ov > v || (ov == v && oj < j)) { v = ov; j = oj; }
}

// ---------------- FPS kernel: one workgroup per batch ----------------
__global__ void __launch_bounds__(FPS_BLOCK)
fps_kernel(const float* __restrict__ abq,
           const unsigned char* __restrict__ mask,
           const int* __restrict__ rand_start,
           int* __restrict__ qidx,
           int n, int m) {
  const int b    = blockIdx.x;
  const int tid  = threadIdx.x;
  const int wave = tid >> 5;
  const int lane = tid & 31;

  __shared__ float4 rowbuf[MAX_N];   // one row of abq_pairs: n x 4 floats (32 KB)
  __shared__ float  redV[2][32];     // ping-pong wave partials (step parity)
  __shared__ int    redJ[2][32];
  __shared__ int    sFar;

  const unsigned char* mb = mask + (size_t)b * n;

  // initial farthest: k-th valid atom, k = rand_start % counts (matches FPSindices)
  if (tid == 0) {
    int counts = 0;
#pragma clang loop unroll(disable)
    for (int j = 0; j < n; ++j) counts += mb[j] ? 1 : 0;
    int k = rand_start[b] % counts;
    int cnt = 0, far = 0;
#pragma clang loop unroll(disable)
    for (int j = 0; j < n; ++j) {
      if (mb[j]) { if (cnt == k) { far = j; break; } ++cnt; }
    }
    sFar = far;
  }
  __syncthreads();
  int farthest = sFar;

  // per-thread column state kept in registers (n <= 2048, 1024 threads -> 2 cols)
  const int CPT = 2;
  float dist[CPT];
  bool  valid[CPT];
  for (int c2 = 0; c2 < CPT; ++c2) {
    int j = tid + c2 * FPS_BLOCK;
    valid[c2] = (j < n) ? (mb[j] != 0) : false;
    dist[c2]  = 1e8f;
  }

  const float* abq_b      = abq + (size_t)b * n * n * 4;
  const uint32_t rowBytes = (uint32_t)n * 16u;                   // n * d * 4
  const uint32_t lds_base = (uint32_t)(uintptr_t)(&rowbuf[0]);   // low 32 bits = LDS offset

  for (int step = 0; step < m; ++step) {
    if (tid == 0) qidx[(size_t)b * m + step] = farthest;
    const int p = step & 1;

    // async-stage row abq[b, farthest, :, :] into LDS; each lane stages exactly
    // the bytes it will itself consume -> s_wait_asynccnt suffices, no barrier.
    uint32_t rbase = (uint32_t)farthest * rowBytes;
    for (uint32_t off = (uint32_t)tid * 16u; off < rowBytes; off += (uint32_t)FPS_BLOCK * 16u) {
      async_copy_b128(lds_base + off, rbase + off, (const void*)abq_b);
    }
    wait_async0();

    // d = ||x||, mask -> -100, distances = min(distances, d); local argmax
    float bv = -3.0e38f; int bj = 0x7fffffff;
    for (int c2 = 0; c2 < CPT; ++c2) {
      int j = tid + c2 * FPS_BLOCK;
      float4 x = rowbuf[j];
      float dd = sqrtf(x.x * x.x + x.y * x.y + x.z * x.z + x.w * x.w);
      dd = valid[c2] ? dd : -100.0f;
      float nd = fminf(dist[c2], dd);
      dist[c2] = nd;
      if (nd > bv) { bv = nd; bj = j; }   // j strictly increasing -> first-index ties
    }

    // wave32 argmax reduce (max value, then min index: matches jnp.argmax)
    for (int off2 = 16; off2 > 0; off2 >>= 1) {
      float ov = __shfl_down(bv, off2, 32);
      int   oj = __shfl_down(bj, off2, 32);
      combine_max_first(bv, bj, ov, oj);
    }
    if (lane == 0) { redV[p][wave] = bv; redJ[p][wave] = bj; }
    __syncthreads();   // single barrier per step (ping-pong removes WAR across steps)

    // every wave redundantly reduces the 32 wave-partials, then lane-0 broadcast
    float v2 = redV[p][lane];
    int   j2 = redJ[p][lane];
    for (int off2 = 16; off2 > 0; off2 >>= 1) {
      float ov = __shfl_down(v2, off2, 32);
      int   oj = __shfl_down(j2, off2, 32);
      combine_max_first(v2, j2, ov, oj);
    }
    farthest = __shfl(j2, 0, 32);
  }
}

// ---------------- gather kernels ----------------
template <int CH>
__global__ void gather_pairs_kernel(const float* __restrict__ src,
                                    const int* __restrict__ qidx,
                                    float* __restrict__ dst,
                                    int n, int m, long long total) {
  long long t = (long long)blockIdx.x * blockDim.x + threadIdx.x;
  if (t >= total) return;
  int qj = (int)(t % m);
  long long tmp = t / m;
  int qi = (int)(tmp % m);
  int b  = (int)(tmp / m);
  int si = qidx[(size_t)b * m + qi];
  int sj = qidx[(size_t)b * m + qj];
  const float* s = src + (((size_t)b * n + si) * n + sj) * CH;
  float*       o = dst + (((size_t)b * m + qi) * m + qj) * CH;
#pragma unroll
  for (int k = 0; k < CH; ++k) o[k] = s[k];
}

__global__ void gather_pairs_generic(const float* __restrict__ src,
                                     const int* __restrict__ qidx,
                                     float* __restrict__ dst,
                                     int n, int m, int ch, long long total) {
  long long t = (long long)blockIdx.x * blockDim.x + threadIdx.x;
  if (t >= total) return;
  int qj = (int)(t % m);
  long long tmp = t / m;
  int qi = (int)(tmp % m);
  int b  = (int)(tmp / m);
  int si = qidx[(size_t)b * m + qi];
  int sj = qidx[(size_t)b * m + qj];
  const float* s = src + (((size_t)b * n + si) * n + sj) * ch;
  float*       o = dst + (((size_t)b * m + qi) * m + qj) * ch;
  for (int k = 0; k < ch; ++k) o[k] = s[k];
}

__global__ void gather_vals_kernel(const float* __restrict__ vals,
                                   const int* __restrict__ qidx,
                                   float* __restrict__ dst,
                                   int n, int c, int m, int total) {
  int t = blockIdx.x * blockDim.x + threadIdx.x;
  if (t >= total) return;
  int cc  = t % c;
  int tmp = t / c;
  int qi  = tmp % m;
  int b   = tmp / m;
  int si  = qidx[(size_t)b * m + qi];
  dst[t] = vals[((size_t)b * n + si) * c + cc];
}

__global__ void gather_mask_kernel(const unsigned char* __restrict__ mask,
                                   const int* __restrict__ qidx,
                                   float* __restrict__ dst,
                                   int n, int m, int total) {
  int t = blockIdx.x * blockDim.x + threadIdx.x;
  if (t >= total) return;
  int qi = t % m;
  int b  = t / m;
  int si = qidx[(size_t)b * m + qi];
  dst[t] = mask[(size_t)b * n + si] ? 1.0f : 0.0f;
}

// ---------------- launch ----------------
extern "C" void kernel_launch(void* const* d_in, const int* in_sizes, int n_in,
                              void* d_out, int out_size, void* d_ws, size_t ws_size,
                              hipStream_t stream) {
  const float*         abq        = (const float*)d_in[0];
  const float*         vals       = (const float*)d_in[1];
  const unsigned char* mask       = (const unsigned char*)d_in[2];
  const float*         edges      = (const float*)d_in[3];
  const int*           rand_start = (const int*)d_in[4];

  const int bs = in_sizes[4];                        // 4
  const int n  = in_sizes[2] / bs;                   // 2048
  const int c  = in_sizes[1] / (bs * n);             // 64
  const long long nn = (long long)n * n;
  const int d  = (int)((long long)in_sizes[0] / (bs * nn));  // 4
  const int e  = (int)((long long)in_sizes[3] / (bs * nn));  // 2
  const int m  = n / 4;                              // round(0.25*n) = 512

  int* qidx = (int*)d_ws;                            // bs*m ints

  // 1) FPS: serial chain, one workgroup per batch, async row staging into LDS
  fps_kernel<<<bs, FPS_BLOCK, 0, stream>>>(abq, mask, rand_start, qidx, n, m);

  // 2) gathers
  float* out_abq   = (float*)d_out;
  float* out_vals  = out_abq  + (size_t)bs * m * m * d;
  float* out_mask  = out_vals + (size_t)bs * m * c;
  float* out_edges = out_mask + (size_t)bs * m;

  const long long totalP = (long long)bs * m * m;
  const int gP = (int)((totalP + 255) / 256);

  if (d == 4)
    gather_pairs_kernel<4><<<gP, 256, 0, stream>>>(abq, qidx, out_abq, n, m, totalP);
  else
    gather_pairs_generic<<<gP, 256, 0, stream>>>(abq, qidx, out_abq, n, m, d, totalP);

  if (e == 2)
    gather_pairs_kernel<2><<<gP, 256, 0, stream>>>(edges, qidx, out_edges, n, m, totalP);
  else
    gather_pairs_generic<<<gP, 256, 0, stream>>>(edges, qidx, out_edges, n, m, e, totalP);

  const int totalV = bs * m * c;
  gather_vals_kernel<<<(totalV + 255) / 256, 256, 0, stream>>>(vals, qidx, out_vals, n, c, m, totalV);

  const int totalM = bs * m;
  gather_mask_kernel<<<(totalM + 255) / 256, 256, 0, stream>>>(mask, qidx, out_mask, n, m, totalM);
}